// Topkpool_3977139716802
// MI455X (gfx1250) — compile-verified
//
#include <hip/hip_runtime.h>
#include <hip/hip_bf16.h>

// Topkpool (DGCNN-style GCN + SortPooling) for MI455X / gfx1250.
// GEMMs use V_WMMA_F32_16X16X4_F32 (fp32 in/out, preserves reference fp32
// semantics; GEMM is ~10 GFLOP total and never the bottleneck — the edge
// scatter-add (~1.7 GB traffic/layer) is).

#define DF    128
#define KPOOL 3
#define NCLS  10
#define BN_EPS 1e-5f

typedef __attribute__((ext_vector_type(2))) float v2f;
typedef __attribute__((ext_vector_type(8))) float v8f;

// ---------------------------------------------------------------- utilities
__global__ void zero_kernel(float* __restrict__ p, long n) {
  long i = (long)blockIdx.x * blockDim.x + threadIdx.x;
  if (i < n) p[i] = 0.0f;
}

__global__ void degree_count(const int* __restrict__ src, const int* __restrict__ dst,
                             float* __restrict__ degO, float* __restrict__ degI, int E) {
  int e = blockIdx.x * blockDim.x + threadIdx.x;
  if (e < E) {
    atomicAdd(&degO[src[e]], 1.0f);
    atomicAdd(&degI[dst[e]], 1.0f);
  }
}

// in-place: deg -> rsqrt(max(deg,1))
__global__ void degree_finalize(float* __restrict__ degO, float* __restrict__ degI, int n) {
  int i = blockIdx.x * blockDim.x + threadIdx.x;
  if (i < n) {
    degO[i] = rsqrtf(fmaxf(degO[i], 1.0f));
    degI[i] = rsqrtf(fmaxf(degI[i], 1.0f));
  }
}

// out[i,d] = h[i,d] * r[i]
__global__ void scale_rows(const float* __restrict__ h, const float* __restrict__ r,
                           float* __restrict__ out, long nd) {
  long i = (long)blockIdx.x * blockDim.x + threadIdx.x;
  if (i < nd) out[i] = h[i] * r[i >> 7];
}

// warp-per-edge: agg[dst[e],:] += hs[src[e],:]   (float4 gather, f32 atomics)
__global__ void scatter_edges(const float* __restrict__ hs, const int* __restrict__ src,
                              const int* __restrict__ dst, float* __restrict__ agg, int E) {
  long tid = (long)blockIdx.x * blockDim.x + threadIdx.x;
  int e    = (int)(tid >> 5);
  int lane = (int)(tid & 31);
  if (e >= E) return;
  int s = src[e], d = dst[e];
  const float4* in = (const float4*)(hs + (size_t)s * DF);
  float4 v = in[lane];                       // 32 lanes x 4 = 128 features
  float* out = agg + (size_t)d * DF + lane * 4;
  atomicAdd(out + 0, v.x);
  atomicAdd(out + 1, v.y);
  atomicAdd(out + 2, v.z);
  atomicAdd(out + 3, v.w);
}

// ------------------------------------------------------------ WMMA fp32 GEMM
// out[N,128] = diag(rscale) * A[N,128] @ W[128,128] + bias
// Block = 256 threads = 8 waves; block covers 16 rows x 128 cols,
// wave w covers cols [16w,16w+16). K loop: 32x V_WMMA_F32_16X16X4_F32.
__global__ __launch_bounds__(256)
void gemm_rowscale_bias(const float* __restrict__ A, const float* __restrict__ rscale,
                        const float* __restrict__ W, const float* __restrict__ bias,
                        float* __restrict__ out, int nrows) {
  int wave = threadIdx.x >> 5;
  int lane = threadIdx.x & 31;
  int half = lane >> 4;          // 0: lanes 0-15, 1: lanes 16-31
  int lh   = lane & 15;
  int row0 = blockIdx.x * 16;
  int n0   = wave * 16;

  int r  = row0 + lh;
  int rc = (r < nrows) ? r : (nrows - 1);       // clamp (N is multiple of 16 anyway)
  float s = rscale[rc];
  const float* arow = A + (size_t)rc * DF + 2 * half;   // K = k+2h, k+2h+1
  const float* wcol = W + (size_t)(2 * half) * DF + n0 + lh;

  v8f c = {0.f, 0.f, 0.f, 0.f, 0.f, 0.f, 0.f, 0.f};
#pragma unroll
  for (int k = 0; k < DF; k += 4) {
    v2f a, b;
    a.x = arow[k]     * s;                 // A[r][k+2h]
    a.y = arow[k + 1] * s;                 // A[r][k+2h+1]
    b.x = wcol[(size_t)k * DF];            // W[k+2h  ][n0+lh]
    b.y = wcol[(size_t)k * DF + DF];       // W[k+2h+1][n0+lh]
    c = __builtin_amdgcn_wmma_f32_16x16x4_f32(false, a, false, b, (short)0, c,
                                              false, false);
  }
  float bv = bias[n0 + lh];
#pragma unroll
  for (int vi = 0; vi < 8; ++vi) {
    int m = row0 + vi + 8 * half;          // D layout: M = vgpr + 8*half, N = lh
    if (m < nrows) out[(size_t)m * DF + n0 + lh] = c[vi] + bv;
  }
}

// --------------------------------------------------------------- batch norm
__global__ __launch_bounds__(DF)
void bn_stats(const float* __restrict__ h, float* __restrict__ sums,
              float* __restrict__ sumsq, int nrows) {
  int c = threadIdx.x;                      // 128 channels, coalesced reads
  float s = 0.f, ss = 0.f;
  for (int r = blockIdx.x; r < nrows; r += gridDim.x) {
    float v = h[(size_t)r * DF + c];
    s += v; ss += v * v;
  }
  atomicAdd(&sums[c], s);
  atomicAdd(&sumsq[c], ss);
}

__global__ void bn_apply_relu(float* __restrict__ h, const float* __restrict__ sums,
                              const float* __restrict__ sumsq, const float* __restrict__ g,
                              const float* __restrict__ bt, int nrows) {
  long i = (long)blockIdx.x * blockDim.x + threadIdx.x;
  long total = (long)nrows * DF;
  if (i >= total) return;
  int c = (int)(i & (DF - 1));
  float invn = 1.0f / (float)nrows;
  float mu  = sums[c] * invn;
  float var = sumsq[c] * invn - mu * mu;
  float x = (h[i] - mu) * rsqrtf(var + BN_EPS) * g[c] + bt[c];
  h[i] = fmaxf(x, 0.0f);
}

// ---------------------------------------------------------------- sort pool
// One block (128 threads) per graph: per-node max, serial top-3 (jax top_k
// tie-breaking: first occurrence), 128-wide bitonic sort per selected row,
// then [384]x[384,10] projection accumulated into score.
#define NPG_MAX 2048
__global__ __launch_bounds__(DF)
void sortpool_kernel(const float* __restrict__ h, const float* __restrict__ Pw,
                     const float* __restrict__ Pb, float* __restrict__ score,
                     const int* __restrict__ pG, const int* __restrict__ pNPG,
                     int init) {
  int G = *pG;
  int g = blockIdx.x;
  if (g >= G) return;
  int npg = *pNPG;
  if (npg > NPG_MAX) npg = NPG_MAX;
  int t = threadIdx.x;

  __shared__ float nodemax[NPG_MAX];
  __shared__ float srt[DF];
  __shared__ float pooled[KPOOL * DF];
  __shared__ int sel[KPOOL];

  for (int n = t; n < npg; n += DF) {
    const float* row = h + ((size_t)g * npg + n) * DF;
    float m = row[0];
    for (int j = 1; j < DF; ++j) m = fmaxf(m, row[j]);
    nodemax[n] = m;
  }
  __syncthreads();

  if (t == 0) {
    for (int k = 0; k < KPOOL; ++k) {
      float best = -3.402823466e38f; int bi = 0;
      for (int n = 0; n < npg; ++n) {
        bool taken = false;
        for (int q = 0; q < k; ++q) taken |= (sel[q] == n);
        float v = nodemax[n];
        if (!taken && v > best) { best = v; bi = n; }
      }
      sel[k] = bi;
    }
  }
  __syncthreads();

  for (int k = 0; k < KPOOL; ++k) {
    const float* row = h + ((size_t)g * npg + sel[k]) * DF;
    srt[t] = row[t];
    __syncthreads();
    for (int sz = 2; sz <= DF; sz <<= 1) {
      for (int j = sz >> 1; j > 0; j >>= 1) {
        int ixj = t ^ j;
        if (ixj > t) {
          bool up = ((t & sz) == 0);
          float a = srt[t], b = srt[ixj];
          if ((a > b) == up) { srt[t] = b; srt[ixj] = a; }
        }
        __syncthreads();
      }
    }
    pooled[k * DF + t] = srt[t];
    __syncthreads();
  }

  if (t < NCLS) {
    float acc = Pb[t];
    for (int j = 0; j < KPOOL * DF; ++j) acc += pooled[j] * Pw[j * NCLS + t];
    float* sp = score + (size_t)g * NCLS + t;
    if (init) *sp = acc; else *sp += acc;
  }
}

// ------------------------------------------------------------------- driver
extern "C" void kernel_launch(void* const* d_in, const int* in_sizes, int n_in,
                              void* d_out, int out_size, void* d_ws, size_t ws_size,
                              hipStream_t stream) {
  const float* x   = (const float*)d_in[0];
  const int*   src = (const int*)d_in[1];
  const int*   dst = (const int*)d_in[2];
  const int*   pG  = (const int*)d_in[3];
  const int*   pNPG= (const int*)d_in[4];
  const float* W[3]  = {(const float*)d_in[5],  (const float*)d_in[9],  (const float*)d_in[13]};
  const float* bb[3] = {(const float*)d_in[6],  (const float*)d_in[10], (const float*)d_in[14]};
  const float* gg[3] = {(const float*)d_in[7],  (const float*)d_in[11], (const float*)d_in[15]};
  const float* bt[3] = {(const float*)d_in[8],  (const float*)d_in[12], (const float*)d_in[16]};
  const float* Pw[4] = {(const float*)d_in[17], (const float*)d_in[19], (const float*)d_in[21], (const float*)d_in[23]};
  const float* Pb[4] = {(const float*)d_in[18], (const float*)d_in[20], (const float*)d_in[22], (const float*)d_in[24]};

  long ND = in_sizes[0];            // N * 128
  int  N  = (int)(ND / DF);
  int  E  = in_sizes[1];
  float* out = (float*)d_out;

  // workspace carve-out (256B aligned)
  char* ws = (char*)d_ws;
  size_t off = 0;
  auto carve = [&](size_t bytes) -> float* {
    float* p = (float*)(ws + off);
    off += (bytes + 255) & ~(size_t)255;
    return p;
  };
  float* degO  = carve((size_t)N * 4);          // -> rsqrt(max(out-deg,1))
  float* degI  = carve((size_t)N * 4);          // -> rsqrt(max(in-deg,1))
  float* stats = carve(256 * 4);                // sums[128] | sumsq[128]
  float* sums  = stats;
  float* sumsq = stats + DF;
  float* B1 = carve((size_t)ND * 4);            // scaled h
  float* B2 = carve((size_t)ND * 4);            // agg
  float* B3 = carve((size_t)ND * 4);            // lin / bn output (next h)
  (void)ws_size; (void)n_in; (void)out_size;

  dim3 blk(256);
  long gN   = (N + 255) / 256;
  long gND  = (ND + 255) / 256;
  long gE   = ((long)E + 255) / 256;
  long gE32 = (((long)E << 5) + 255) / 256;

  // degrees
  zero_kernel<<<dim3((unsigned)gN), blk, 0, stream>>>(degO, N);
  zero_kernel<<<dim3((unsigned)gN), blk, 0, stream>>>(degI, N);
  degree_count<<<dim3((unsigned)gE), blk, 0, stream>>>(src, dst, degO, degI, E);
  degree_finalize<<<dim3((unsigned)gN), blk, 0, stream>>>(degO, degI, N);

  // rep 0: sort-pool the raw features (initializes score)
  sortpool_kernel<<<dim3((unsigned)N), dim3(DF), 0, stream>>>(x, Pw[0], Pb[0], out, pG, pNPG, 1);

  const float* hin = x;
  for (int l = 0; l < 3; ++l) {
    scale_rows<<<dim3((unsigned)gND), blk, 0, stream>>>(hin, degO, B1, ND);
    zero_kernel<<<dim3((unsigned)gND), blk, 0, stream>>>(B2, ND);
    scatter_edges<<<dim3((unsigned)gE32), blk, 0, stream>>>(B1, src, dst, B2, E);
    zero_kernel<<<dim3(1), blk, 0, stream>>>(stats, 256);
    gemm_rowscale_bias<<<dim3((unsigned)((N + 15) / 16)), blk, 0, stream>>>(
        B2, degI, W[l], bb[l], B3, N);
    bn_stats<<<dim3(512), dim3(DF), 0, stream>>>(B3, sums, sumsq, N);
    bn_apply_relu<<<dim3((unsigned)gND), blk, 0, stream>>>(B3, sums, sumsq, gg[l], bt[l], N);
    sortpool_kernel<<<dim3((unsigned)N), dim3(DF), 0, stream>>>(B3, Pw[l + 1], Pb[l + 1], out, pG, pNPG, 0);
    hin = B3;
  }
}